// Hypernym_61624190763537
// MI455X (gfx1250) — compile-verified
//
#include <hip/hip_runtime.h>
#include <stdint.h>

// EmbeddingBag(sum): out[b,:] = sum_l w[b,l] * emb[idx[b,l], :]
// B=8192, L=50, V=100000, D=300.
// Memory-bound gather: ~135MB HBM minimum (~6us at 23.3TB/s), 491MB of L2
// gather traffic; the 120MB table is L2-resident (192MB L2). CDNA5 async
// global->LDS copies (ASYNCcnt) put the whole 60KB per-sample row set in
// flight without VGPR pressure, then the weighted reduce reads LDS.

#define BB 8192
#define LL 50
#define DD 300
#define ROW_BYTES (DD * 4)                 // 1200 B per table row
#define CHUNKS_PER_ROW 75                  // D/4 float4 chunks per row
#define THREADS 96                         // 3 waves; lanes 0..74 do the work

typedef float v4f __attribute__((ext_vector_type(4)));

__global__ __launch_bounds__(THREADS)
void hypernym_embbag_kernel(const int* __restrict__ idx,
                            const float* __restrict__ w,
                            const float* __restrict__ emb,
                            float* __restrict__ out)
{
    __shared__ __align__(16) float rows[LL * DD];   // 60000 B gathered rows
    __shared__ float    wlds[LL];
    __shared__ uint32_t offlds[LL];                 // pre-scaled row byte offsets

    const int b = blockIdx.x;
    const int t = threadIdx.x;

    // ---- stage 0: stage weights + pre-scaled row offsets (idx*1200) ----
    if (t < LL) {
        offlds[t] = (uint32_t)idx[b * LL + t] * (uint32_t)ROW_BYTES;
        wlds[t]   = __builtin_nontemporal_load(&w[b * LL + t]);
    }
    __syncthreads();

    // ---- stage 1: async gather 50 rows x 1200B into LDS ----
    // Lane t owns the fixed 16B column chunk t of every row; per-row address
    // is just base + offlds[r] (no div/mod, no per-lane multiplies).
    if (t < CHUNKS_PER_ROW) {
        uint32_t       loff    = (uint32_t)(uintptr_t)(&rows[0]) + (uint32_t)(t << 4);
        const uint64_t colbase = (uint64_t)(uintptr_t)emb + (uint64_t)(t << 4);
        #pragma unroll 5
        for (int r = 0; r < LL; ++r) {
            const uint64_t src = colbase + (uint64_t)offlds[r];
            asm volatile("global_load_async_to_lds_b128 %0, %1, off"
                         :: "v"(loff), "v"(src)
                         : "memory");
            loff += ROW_BYTES;
        }
    }
    // Wait for this wave's async copies, then barrier so every wave's LDS
    // writes are visible before the reduction.
    asm volatile("s_wait_asynccnt 0" ::: "memory");
    __syncthreads();

    // ---- stage 2: weighted reduce, one float4 column slice per lane ----
    if (t < CHUNKS_PER_ROW) {
        const v4f* rows4 = (const v4f*)rows;
        v4f acc = (v4f)(0.0f);
        #pragma unroll 10
        for (int l = 0; l < LL; ++l) {
            const float wl = wlds[l];
            const v4f   v  = rows4[l * CHUNKS_PER_ROW + t];
            acc += wl * v;
        }
        v4f* o4 = (v4f*)(out + (size_t)b * DD);
        __builtin_nontemporal_store(acc, &o4[t]);   // out is write-once: keep L2 for the table
    }
}

extern "C" void kernel_launch(void* const* d_in, const int* in_sizes, int n_in,
                              void* d_out, int out_size, void* d_ws, size_t ws_size,
                              hipStream_t stream) {
    const int*   idx = (const int*)d_in[0];    // hnym_idx     [B, L]
    const float* w   = (const float*)d_in[1];  // hnym_weights [B, L]
    const float* emb = (const float*)d_in[2];  // emb_table    [V, D]
    float* out = (float*)d_out;                // [B, D]
    (void)in_sizes; (void)n_in; (void)out_size; (void)d_ws; (void)ws_size;

    hypernym_embbag_kernel<<<dim3(BB), dim3(THREADS), 0, stream>>>(idx, w, emb, out);
}